// LightGraphSAGE_26053271617569
// MI455X (gfx1250) — compile-verified
//
#include <hip/hip_runtime.h>
#include <hip/hip_bf16.h>

// ---------------------------------------------------------------------------
// LightGraphSAGE on MI455X (gfx1250).
// Dense GEMMs use V_WMMA_F32_16X16X4_F32 (wave32, fp32 in/out) since the
// workload is bandwidth-bound on the edge scatter (~2 GB HBM traffic) and
// the GEMMs total only ~5 GFLOP; fp32 WMMA preserves reference precision.
// ---------------------------------------------------------------------------

typedef __attribute__((ext_vector_type(2))) float v2f;
typedef __attribute__((ext_vector_type(8))) float v8f;

#define HID 64
#define N_NODES_C 100000
#define N_EDGES_C 3200000LL
#define BN_EPS_C 1e-5f

__device__ __forceinline__ v8f wmma_f32(v2f a, v2f b, v8f c) {
  // 8 args: (neg_a, A, neg_b, B, c_mod, C, reuse_a, reuse_b)
  return __builtin_amdgcn_wmma_f32_16x16x4_f32(
      false, a, false, b, (short)0, c, false, false);
}

// ---------------------------------------------------------------------------
// GEMM: Out[M,64] = A[M,K] @ W[K,64] + bias[64]         (K = 128 here)
// One wave -> one 16x16 tile; block (256 thr = 8 waves) -> 32 rows x 64 cols.
// M % 32 == 0 (100000/32 = 3125), so no bounds checks; EXEC stays all-ones.
// ---------------------------------------------------------------------------
__global__ void gemm_proj_kernel(const float* __restrict__ A,
                                 const float* __restrict__ W,
                                 const float* __restrict__ bias,
                                 float* __restrict__ Out,
                                 int M, int K) {
  const int N = HID;
  int wave = threadIdx.x >> 5;          // 0..7
  int lane = threadIdx.x & 31;
  int rt = wave >> 2;                   // row tile 0..1
  int ct = wave & 3;                    // col tile 0..3
  int m0 = blockIdx.x * 32 + rt * 16;
  int n0 = ct * 16;
  int half = lane >> 4;                 // 0 or 1 -> K offset pair
  int l16 = lane & 15;
  int m = m0 + l16;                     // A row for this lane
  int n = n0 + l16;                     // B/C col for this lane

  v8f c = {};
  for (int k0 = 0; k0 < K; k0 += 4) {
    int k = k0 + half * 2;
    v2f a, b;
    a.x = A[m * K + k];
    a.y = A[m * K + k + 1];
    b.x = W[k * N + n];
    b.y = W[(k + 1) * N + n];
    c = wmma_f32(a, b, c);
  }
  float bv = bias[n];
#pragma unroll
  for (int v = 0; v < 8; ++v) {
    int mr = m0 + v + half * 8;         // C/D layout: m = vgpr + 8*(lane>>4)
    Out[mr * N + n] = c[v] + bv;
  }
}

// ---------------------------------------------------------------------------
// SAGE GEMM: Out = Aggr @ Wl + H @ Wr + bias   (all [*,64] x [64,64])
// ---------------------------------------------------------------------------
__global__ void gemm_sage_kernel(const float* __restrict__ Aggr,
                                 const float* __restrict__ H,
                                 const float* __restrict__ Wl,
                                 const float* __restrict__ Wr,
                                 const float* __restrict__ bias,
                                 float* __restrict__ Out, int M) {
  const int N = HID, K = HID;
  int wave = threadIdx.x >> 5;
  int lane = threadIdx.x & 31;
  int rt = wave >> 2;
  int ct = wave & 3;
  int m0 = blockIdx.x * 32 + rt * 16;
  int n0 = ct * 16;
  int half = lane >> 4;
  int l16 = lane & 15;
  int m = m0 + l16;
  int n = n0 + l16;

  v8f c = {};
  for (int k0 = 0; k0 < K; k0 += 4) {     // aggregated-neighbor path
    int k = k0 + half * 2;
    v2f a, b;
    a.x = Aggr[m * K + k];
    a.y = Aggr[m * K + k + 1];
    b.x = Wl[k * N + n];
    b.y = Wl[(k + 1) * N + n];
    c = wmma_f32(a, b, c);
  }
  for (int k0 = 0; k0 < K; k0 += 4) {     // self path
    int k = k0 + half * 2;
    v2f a, b;
    a.x = H[m * K + k];
    a.y = H[m * K + k + 1];
    b.x = Wr[k * N + n];
    b.y = Wr[(k + 1) * N + n];
    c = wmma_f32(a, b, c);
  }
  float bv = bias[n];
#pragma unroll
  for (int v = 0; v < 8; ++v) {
    int mr = m0 + v + half * 8;
    Out[mr * N + n] = c[v] + bv;
  }
}

// ---------------------------------------------------------------------------
// Edge scatter: one thread per (edge, channel). summed[dst,c] += h[src,c];
// lane c==0 also bumps the in-degree counter.  edge_index is int64 [2, E].
// ---------------------------------------------------------------------------
__global__ void edge_scatter_kernel(const long long* __restrict__ ei,
                                    const float* __restrict__ H,
                                    float* __restrict__ summed,
                                    float* __restrict__ cnt,
                                    long long nEdges) {
  long long gid = (long long)blockIdx.x * blockDim.x + threadIdx.x;
  long long e = gid >> 6;
  int c = (int)(gid & 63);
  if (e >= nEdges) return;
  long long src = ei[e];
  long long dst = ei[nEdges + e];
  float v = H[src * HID + c];
  atomicAdd(&summed[dst * HID + c], v);
  if (c == 0) atomicAdd(&cnt[dst], 1.0f);
}

// aggr[i,c] = summed[i,c] / max(cnt[i], 1)     (in place)
__global__ void aggr_norm_kernel(float* __restrict__ summed,
                                 const float* __restrict__ cnt, int M) {
  int gid = blockIdx.x * blockDim.x + threadIdx.x;
  if (gid >= M * HID) return;
  int row = gid >> 6;
  float d = fmaxf(cnt[row], 1.0f);
  summed[gid] = summed[gid] / d;
}

// ---------------------------------------------------------------------------
// BN stats: per-channel sum -> stats[0..63], sum of squares -> stats[64..127]
// ---------------------------------------------------------------------------
__global__ void bn_stats_kernel(const float* __restrict__ P,
                                float* __restrict__ stats, int M) {
  __shared__ float ssum[256];
  __shared__ float ssq[256];
  int c = threadIdx.x & 63;
  int rlo = threadIdx.x >> 6;           // 0..3 rows per block-iteration
  float s = 0.f, q = 0.f;
  for (int m = blockIdx.x * 4 + rlo; m < M; m += gridDim.x * 4) {
    float v = P[m * HID + c];
    s += v;
    q += v * v;
  }
  ssum[threadIdx.x] = s;
  ssq[threadIdx.x] = q;
  __syncthreads();
  if (threadIdx.x < 64) {
    float S = ssum[threadIdx.x] + ssum[threadIdx.x + 64] +
              ssum[threadIdx.x + 128] + ssum[threadIdx.x + 192];
    float Q = ssq[threadIdx.x] + ssq[threadIdx.x + 64] +
              ssq[threadIdx.x + 128] + ssq[threadIdx.x + 192];
    atomicAdd(&stats[threadIdx.x], S);
    atomicAdd(&stats[64 + threadIdx.x], Q);
  }
}

// Out = relu(g * (P - mean) * rsqrt(var + eps) + b) [+ resid]
__global__ void bn_apply_kernel(const float* __restrict__ P,
                                const float* __restrict__ stats,
                                const float* __restrict__ g,
                                const float* __restrict__ b,
                                const float* __restrict__ resid,
                                float* __restrict__ Out, int M) {
  int gid = blockIdx.x * blockDim.x + threadIdx.x;
  if (gid >= M * HID) return;
  int c = gid & 63;
  float invM = 1.0f / (float)M;
  float mean = stats[c] * invM;
  float var = stats[64 + c] * invM - mean * mean;   // biased, as torch/jnp
  float inv = rsqrtf(var + BN_EPS_C);
  float v = (P[gid] - mean) * inv * g[c] + b[c];
  v = fmaxf(v, 0.0f);
  if (resid) v += resid[gid];
  Out[gid] = v;
}

// Output head: out[m, 0..1] = H[m,:] @ Wo + bo   (N=2: scalar dot products)
__global__ void out_gemm_kernel(const float* __restrict__ H,
                                const float* __restrict__ Wo,
                                const float* __restrict__ bo,
                                float* __restrict__ out, int M) {
  int m = blockIdx.x * blockDim.x + threadIdx.x;
  if (m >= M) return;
  float a0 = bo[0], a1 = bo[1];
#pragma unroll 8
  for (int k = 0; k < HID; ++k) {
    float h = H[m * HID + k];
    a0 += h * Wo[k * 2 + 0];
    a1 += h * Wo[k * 2 + 1];
  }
  out[m * 2 + 0] = a0;
  out[m * 2 + 1] = a1;
}

// ---------------------------------------------------------------------------
extern "C" void kernel_launch(void* const* d_in, const int* in_sizes, int n_in,
                              void* d_out, int out_size, void* d_ws, size_t ws_size,
                              hipStream_t stream) {
  (void)in_sizes; (void)n_in; (void)out_size; (void)ws_size;

  const float* x      = (const float*)d_in[0];        // [100000,128]
  const long long* ei = (const long long*)d_in[1];    // [2, 3.2M] int64
  const float* Wp  = (const float*)d_in[2];
  const float* bp  = (const float*)d_in[3];
  const float* g_in = (const float*)d_in[4];
  const float* b_in = (const float*)d_in[5];
  const float* Wl0 = (const float*)d_in[6];
  const float* bl0 = (const float*)d_in[7];
  const float* Wr0 = (const float*)d_in[8];
  const float* g0  = (const float*)d_in[9];
  const float* b0  = (const float*)d_in[10];
  const float* Wl1 = (const float*)d_in[11];
  const float* bl1 = (const float*)d_in[12];
  const float* Wr1 = (const float*)d_in[13];
  const float* g1  = (const float*)d_in[14];
  const float* b1  = (const float*)d_in[15];
  const float* Wo  = (const float*)d_in[16];
  const float* bo  = (const float*)d_in[17];
  float* out = (float*)d_out;

  const int M = N_NODES_C;
  const long long E = N_EDGES_C;
  const size_t feat = (size_t)M * HID;

  float* bufA  = (float*)d_ws;          // h0 (post-proj activations)
  float* bufB  = bufA + feat;           // h1
  float* bufC  = bufB + feat;           // summed/aggr, then h2
  float* bufD  = bufC + feat;           // pre-BN GEMM output
  float* cnt   = bufD + feat;           // in-degree [M]
  float* stats = cnt + M;               // [128] channel sums / sumsq

  const int elemBlocks = (M * HID + 255) / 256;
  const int edgeBlocks = (int)((E * 64 + 255) / 256);

  // ---- input projection + BN + relu -> bufA (h0) ----
  gemm_proj_kernel<<<M / 32, 256, 0, stream>>>(x, Wp, bp, bufD, M, 128);
  hipMemsetAsync(stats, 0, 128 * sizeof(float), stream);
  bn_stats_kernel<<<512, 256, 0, stream>>>(bufD, stats, M);
  bn_apply_kernel<<<elemBlocks, 256, 0, stream>>>(bufD, stats, g_in, b_in,
                                                  nullptr, bufA, M);

  // ---- layer 0: aggregate h0, SAGE GEMM, BN+relu -> bufB (h1) ----
  hipMemsetAsync(bufC, 0, feat * sizeof(float), stream);
  hipMemsetAsync(cnt, 0, M * sizeof(float), stream);
  edge_scatter_kernel<<<edgeBlocks, 256, 0, stream>>>(ei, bufA, bufC, cnt, E);
  aggr_norm_kernel<<<elemBlocks, 256, 0, stream>>>(bufC, cnt, M);
  gemm_sage_kernel<<<M / 32, 256, 0, stream>>>(bufC, bufA, Wl0, Wr0, bl0,
                                               bufD, M);
  hipMemsetAsync(stats, 0, 128 * sizeof(float), stream);
  bn_stats_kernel<<<512, 256, 0, stream>>>(bufD, stats, M);
  bn_apply_kernel<<<elemBlocks, 256, 0, stream>>>(bufD, stats, g0, b0,
                                                  nullptr, bufB, M);

  // ---- layer 1: aggregate h1, SAGE GEMM, BN+relu + residual(h0) -> bufC ----
  hipMemsetAsync(bufC, 0, feat * sizeof(float), stream);
  hipMemsetAsync(cnt, 0, M * sizeof(float), stream);
  edge_scatter_kernel<<<edgeBlocks, 256, 0, stream>>>(ei, bufB, bufC, cnt, E);
  aggr_norm_kernel<<<elemBlocks, 256, 0, stream>>>(bufC, cnt, M);
  gemm_sage_kernel<<<M / 32, 256, 0, stream>>>(bufC, bufB, Wl1, Wr1, bl1,
                                               bufD, M);
  hipMemsetAsync(stats, 0, 128 * sizeof(float), stream);
  bn_stats_kernel<<<512, 256, 0, stream>>>(bufD, stats, M);
  bn_apply_kernel<<<elemBlocks, 256, 0, stream>>>(bufD, stats, g1, b1,
                                                  bufA, bufC, M);   // + h0

  // ---- output head ----
  out_gemm_kernel<<<(M + 255) / 256, 256, 0, stream>>>(bufC, Wo, bo, out, M);
}